// StateUpdate_91122026152397
// MI455X (gfx1250) — compile-verified
//
#include <hip/hip_runtime.h>
#include <hip/hip_bf16.h>
#include <math.h>

// CDNA5 / gfx1250 wave32 WMMA implementation of the GNN state update.
// One workgroup (256 threads = 8 wave32) per node. All MLP GEMMs use
// V_WMMA_F32_16X16X4_F32 (exact f32) with LDS-staged, fragment-swizzled B.
// p_nn is staged to LDS during edge-feature build so it is read from HBM
// exactly once; q_nn -> X_e uses async global->LDS copies when available.

typedef __attribute__((ext_vector_type(2))) float v2f;
typedef __attribute__((ext_vector_type(8))) float v8f;

#define NSZ 32
#define NBR 64
#define EFP 196          // edge features padded 193 -> 196 (K multiple of 4)
#define H1S 132          // H1/H2 row stride (132 mod 64 banks = 4)
#define SDK 1.7320508075688772f   // sqrt(NK=3)

// ---- shared memory layout (float offsets) ----
#define OFF_W1   0                         // 49 ksteps * 8 ntiles * 64 = 25088
#define OFF_B1   (OFF_W1 + 49*8*64)        // 128
#define OFF_W2E  (OFF_B1 + 128)            // eqkm W1 frags: 8*2*64 = 1024
#define OFF_W2P  (OFF_W2E + 1024)          // epkm W1 frags: 1024
#define OFF_W2V  (OFF_W2P + 1024)          // evm  W1 frags: 16*4*64 = 4096
#define OFF_B2   (OFF_W2V + 4096)          // 128
#define OFF_W3E  (OFF_B2 + 128)            // eqkm W2 frags: 8*1*64 = 512
#define OFF_W3P  (OFF_W3E + 512)           // epkm W2 frags: 512
#define OFF_W3V  (OFF_W3P + 512)           // evm  W2 frags: 4096
#define OFF_B3   (OFF_W3V + 4096)          // 96 (16 pad | 16 pad | 64)
#define OFF_XE   (OFF_B3 + 96)             // 64 * 196 = 12544
#define OFF_H1   (OFF_XE + 64*EFP)         // 64 * 132 = 8448
#define OFF_H2   (OFF_H1 + 64*H1S)         // 8448
#define OFF_KQ   (OFF_H2 + 64*H1S)         // 64*16
#define OFF_EPK  (OFF_KQ + 64*16)          // 64*16
#define OFF_V    (OFF_EPK + 64*16)         // 64*68 (stride 68: 68%64=4)
#define OFF_XN   (OFF_V + 64*68)           // 64
#define OFF_PS   (OFF_XN + 64)             // 96  (p)
#define OFF_RS   (OFF_PS + 96)             // 192 (r_nn)
#define OFF_DS   (OFF_RS + 192)            // 64  (d_nn)
#define OFF_TMP  (OFF_DS + 64)             // 64  (scalar MLP scratch)
#define OFF_Q    (OFF_TMP + 64)            // 16  (Q: 12 values)
#define OFF_MQ   (OFF_Q + 16)              // 128
#define OFF_MP   (OFF_MQ + 128)            // 384
#define OFF_MPS  (OFF_MP + 384)            // 4 (per-head middle-block sums)
#define OFF_ZQ   (OFF_MPS + 4)             // 64
#define OFF_ZP   (OFF_ZQ + 64)             // 192
#define OFF_QH   (OFF_ZP + 192)            // 32
#define OFF_PH   (OFF_QH + 32)             // 96
#define OFF_PN   (OFF_PH + 96)             // p_nn staged: 64*3*32 = 6144
#define SM_FLOATS (OFF_PN + 6144)

static_assert(SM_FLOATS * 4 <= 320 * 1024, "LDS over 320KB/WGP");

#if __has_builtin(__builtin_amdgcn_global_load_async_to_lds_b32) && \
    __has_builtin(__builtin_amdgcn_s_wait_asynccnt)
#define HAVE_ASYNC_LDS 1
typedef __attribute__((address_space(1))) int* g_i32p;
typedef __attribute__((address_space(3))) int* l_i32p;
__device__ __forceinline__ void async_g2l_b32(const float* g, float* l) {
  __builtin_amdgcn_global_load_async_to_lds_b32(
      (g_i32p)g, (l_i32p)l, /*offset=*/0, /*cpol=*/0);
}
#else
#define HAVE_ASYNC_LDS 0
#endif

struct Params {
  const float *q, *p, *q_nn, *p_nn, *d_nn, *r_nn;
  const float *nqW0, *nqB0, *nqW1, *nqB1, *nqW2, *nqB2;
  const float *eqW0, *eqB0, *eqW1, *eqB1, *eqW2, *eqB2;
  const float *epW0, *epB0, *epW1, *epB1, *epW2, *epB2;
  const float *evW0, *evB0, *evW1, *evB1, *evW2, *evB2;
  const float *qpW0, *qpB0, *qpW1, *qpB1, *qpW2, *qpB2;
  const float *ppW;
  float *out_q, *out_p;
  int N;
};

__device__ __forceinline__ v8f wmma4(v2f a, v2f b, v8f c) {
  // V_WMMA_F32_16X16X4_F32: D = A(16x4) * B(4x16) + C(16x16), exact f32
  return __builtin_amdgcn_wmma_f32_16x16x4_f32(
      /*neg_a=*/false, a, /*neg_b=*/false, b,
      /*c_mod=*/(short)0, c, /*reuse_a=*/false, /*reuse_b=*/false);
}

// Stage W (actK x actN, row stride ldw) into B-fragment-swizzled LDS:
// frag f = kstep*totNtiles + (ntOff+nt); lane holds its float2 contiguously.
// Lane L, elem v: k = kstep*4 + 2*(L>>4) + v ; n = nt*16 + (L&15).
__device__ __forceinline__ void stage_bfrags(
    float* dst, const float* W, int actK, int actN, int ldw,
    int Ksteps, int ntiles, int totNtiles, int ntOff, int tid, int nth) {
  int total = Ksteps * ntiles * 32;
  for (int i = tid; i < total; i += nth) {
    int lane = i & 31;
    int f    = i >> 5;
    int nt   = f % ntiles;
    int ks   = f / ntiles;
    int k    = ks * 4 + ((lane >> 4) << 1);
    int n    = nt * 16 + (lane & 15);
    float x0 = (k     < actK && n < actN) ? W[k * ldw + n]       : 0.0f;
    float x1 = (k + 1 < actK && n < actN) ? W[(k + 1) * ldw + n] : 0.0f;
    int frag = ks * totNtiles + (ntOff + nt);
    dst[frag * 64 + lane * 2 + 0] = x0;
    dst[frag * 64 + lane * 2 + 1] = x1;
  }
}

__device__ __forceinline__ float lrelu(float x) { return x > 0.0f ? x : 0.01f * x; }

extern "C" __global__ __launch_bounds__(256, 1)
void state_update_kernel(Params P) {
  extern __shared__ float sm[];
  const int tid  = threadIdx.x;
  const int node = blockIdx.x;
  const int lane = tid & 31;
  const int wv   = tid >> 5;

  const float* qg = P.q    + (size_t)node * NSZ;
  const float* pg = P.p    + (size_t)node * 3 * NSZ;
  const float* qn = P.q_nn + (size_t)node * NBR * NSZ;
  const float* pn = P.p_nn + (size_t)node * NBR * 3 * NSZ;
  const float* dn = P.d_nn + (size_t)node * NBR;
  const float* rn = P.r_nn + (size_t)node * NBR * 3;

  // ================= Phase A: stage weights + per-node small tensors ========
  stage_bfrags(sm + OFF_W1,  P.eqW0, 193, 32, 32, 49, 2, 8, 0, tid, 256);
  stage_bfrags(sm + OFF_W1,  P.epW0, 193, 32, 32, 49, 2, 8, 2, tid, 256);
  stage_bfrags(sm + OFF_W1,  P.evW0, 193, 64, 64, 49, 4, 8, 4, tid, 256);
  stage_bfrags(sm + OFF_W2E, P.eqW1, 32, 32, 32,  8, 2, 2, 0, tid, 256);
  stage_bfrags(sm + OFF_W2P, P.epW1, 32, 32, 32,  8, 2, 2, 0, tid, 256);
  stage_bfrags(sm + OFF_W2V, P.evW1, 64, 64, 64, 16, 4, 4, 0, tid, 256);
  stage_bfrags(sm + OFF_W3E, P.eqW2, 32,  3,  3,  8, 1, 1, 0, tid, 256);
  stage_bfrags(sm + OFF_W3P, P.epW2, 32,  9,  9,  8, 1, 1, 0, tid, 256);
  stage_bfrags(sm + OFF_W3V, P.evW2, 64, 64, 64, 16, 4, 4, 0, tid, 256);
  if (tid < 128) {
    sm[OFF_B1 + tid] = (tid < 32) ? P.eqB0[tid] : (tid < 64) ? P.epB0[tid - 32] : P.evB0[tid - 64];
    sm[OFF_B2 + tid] = (tid < 32) ? P.eqB1[tid] : (tid < 64) ? P.epB1[tid - 32] : P.evB1[tid - 64];
  }
  if (tid < 96) {
    float b;
    if (tid < 16)      b = (tid < 3)  ? P.eqB2[tid]      : 0.0f;
    else if (tid < 32) b = (tid < 25) ? P.epB2[tid - 16] : 0.0f;
    else               b = P.evB2[tid - 32];
    sm[OFF_B3 + tid] = b;
  }
  if (tid < 96)  sm[OFF_PS + tid] = pg[tid];
  if (tid < 192) sm[OFF_RS + tid] = rn[tid];
  if (tid < 64)  sm[OFF_DS + tid] = dn[tid];
  if (tid < 32)  sm[OFF_XN + tid] = qg[tid];
  if (tid >= 32 && tid < 64) {            // X_n[32+s] = |p[:,s]|
    int s = tid - 32;
    float a = pg[s], b = pg[32 + s], c = pg[64 + s];
    sm[OFF_XN + tid] = sqrtf(a * a + b * b + c * c + 1e-6f);
  }
  __syncthreads();

  // ===== Phase B: build X_e (64 x 193, pad 196) in LDS; stage p_nn to LDS ===
  for (int i = tid; i < NBR * NSZ; i += 256) {
    int e = i >> 5, s = i & 31;
    float r0 = sm[OFF_RS + e * 3 + 0];
    float r1 = sm[OFF_RS + e * 3 + 1];
    float r2 = sm[OFF_RS + e * 3 + 2];
    const float* pne = pn + (size_t)(e * 3) * NSZ + s;
    float a0 = pne[0], a1 = pne[NSZ], a2 = pne[2 * NSZ];
    float* xe = sm + OFF_XE + e * EFP;
#if HAVE_ASYNC_LDS
    async_g2l_b32(qn + e * NSZ + s, xe + 65 + s);   // ASYNCcnt-tracked copy
#else
    xe[65 + s]  = qn[e * NSZ + s];
#endif
    xe[1 + s]   = sm[OFF_XN + s];
    xe[33 + s]  = sm[OFF_XN + 32 + s];
    xe[97 + s]  = sqrtf(a0 * a0 + a1 * a1 + a2 * a2 + 1e-6f);
    xe[129 + s] = sm[OFF_PS + s] * r0 + sm[OFF_PS + 32 + s] * r1 + sm[OFF_PS + 64 + s] * r2;
    xe[161 + s] = a0 * r0 + a1 * r1 + a2 * r2;
    // stage p_nn for the Zp phase (single HBM read of p_nn overall)
    sm[OFF_PN + (e * 3 + 0) * NSZ + s] = a0;
    sm[OFF_PN + (e * 3 + 1) * NSZ + s] = a1;
    sm[OFF_PN + (e * 3 + 2) * NSZ + s] = a2;
    if (s == 0) xe[0] = sm[OFF_DS + e];
    if (s < 3)  xe[193 + s] = 0.0f;
  }
#if HAVE_ASYNC_LDS
  __builtin_amdgcn_s_wait_asynccnt(0);
#endif
  __syncthreads();

  // ============ Phase C: nqm scalar MLP (tiny) then Layer-1 WMMA ============
  if (tid < 32) {
    float acc = P.nqB0[tid];
    for (int k = 0; k < 64; ++k) acc += sm[OFF_XN + k] * P.nqW0[k * 32 + tid];
    sm[OFF_TMP + tid] = lrelu(acc);
  }
  __syncthreads();
  if (tid < 32) {
    float acc = P.nqB1[tid];
    for (int k = 0; k < 32; ++k) acc += sm[OFF_TMP + k] * P.nqW1[k * 32 + tid];
    sm[OFF_TMP + 32 + tid] = lrelu(acc);
  }
  __syncthreads();
  if (tid < 12) {
    float acc = P.nqB2[tid];
    for (int k = 0; k < 32; ++k) acc += sm[OFF_TMP + 32 + k] * P.nqW2[k * 12 + tid];
    sm[OFF_Q + tid] = acc;               // Q[sel][h][k] = Q[sel*6 + h*3 + k]
  }

  // Layer 1: X_e(64x196) @ W1cat(196x128) -> H1, bias + leaky-relu
  {
    const int arow = lane & 15;
    const int koff = (lane >> 4) << 1;
    v8f c0 = {}, c1 = {}, c2 = {}, c3 = {};
    const v2f* w1 = (const v2f*)(sm + OFF_W1);
    for (int ks = 0; ks < 49; ++ks) {
      v2f b = w1[(ks * 8 + wv) * 32 + lane];
      const float* xb = sm + OFF_XE + ks * 4 + koff;
      v2f a0 = *(const v2f*)(xb + (arow + 0) * EFP);
      v2f a1 = *(const v2f*)(xb + (arow + 16) * EFP);
      v2f a2 = *(const v2f*)(xb + (arow + 32) * EFP);
      v2f a3 = *(const v2f*)(xb + (arow + 48) * EFP);
      c0 = wmma4(a0, b, c0);
      c1 = wmma4(a1, b, c1);
      c2 = wmma4(a2, b, c2);
      c3 = wmma4(a3, b, c3);
    }
    const int cc = wv * 16 + arow;       // D: n = lane&15, m = v + 8*(lane>>4)
    const int rb = (lane >> 4) * 8;
    const float bias = sm[OFF_B1 + cc];
#pragma unroll
    for (int v = 0; v < 8; ++v) {
      sm[OFF_H1 + (rb + v + 0) * H1S + cc]  = lrelu(c0[v] + bias);
      sm[OFF_H1 + (rb + v + 16) * H1S + cc] = lrelu(c1[v] + bias);
      sm[OFF_H1 + (rb + v + 32) * H1S + cc] = lrelu(c2[v] + bias);
      sm[OFF_H1 + (rb + v + 48) * H1S + cc] = lrelu(c3[v] + bias);
    }
  }
  __syncthreads();

  // ================= Phase D: Layer-2 WMMA (32 tiles / 8 waves) =============
  {
    const int arow = lane & 15;
    const int koff = (lane >> 4) << 1;
    for (int it = 0; it < 4; ++it) {
      int t = wv + it * 8;
      int mlp, mt, nt;
      if (t < 8)       { mlp = 0; mt = t >> 1;        nt = t & 1; }
      else if (t < 16) { mlp = 1; int u = t - 8;  mt = u >> 1; nt = u & 1; }
      else             { mlp = 2; int u = t - 16; mt = u >> 2; nt = u & 3; }
      const int Ain = (mlp == 0) ? 0 : (mlp == 1) ? 32 : 64;
      const int Ks  = (mlp == 2) ? 16 : 8;
      const int Bnt = (mlp == 2) ? 4 : 2;
      const v2f* wB = (const v2f*)(sm + ((mlp == 0) ? OFF_W2E : (mlp == 1) ? OFF_W2P : OFF_W2V));
      v8f c = {};
      for (int ks = 0; ks < Ks; ++ks) {
        v2f b = wB[(ks * Bnt + nt) * 32 + lane];
        v2f a = *(const v2f*)(sm + OFF_H1 + (mt * 16 + arow) * H1S + Ain + ks * 4 + koff);
        c = wmma4(a, b, c);
      }
      const int cc = nt * 16 + arow;
      const int rb = (lane >> 4) * 8;
      const float bias = sm[OFF_B2 + Ain + cc];
#pragma unroll
      for (int v = 0; v < 8; ++v)
        sm[OFF_H2 + (mt * 16 + rb + v) * H1S + Ain + cc] = lrelu(c[v] + bias);
    }
  }
  __syncthreads();

  // ================= Phase E: Layer-3 WMMA (24 tiles / 8 waves) =============
  {
    const int arow = lane & 15;
    const int koff = (lane >> 4) << 1;
    for (int it = 0; it < 3; ++it) {
      int t = wv + it * 8;
      int mlp, mt, nt;
      if (t < 4)      { mlp = 0; mt = t;     nt = 0; }
      else if (t < 8) { mlp = 1; mt = t - 4; nt = 0; }
      else            { mlp = 2; int u = t - 8; mt = u >> 2; nt = u & 3; }
      const int Ain = (mlp == 0) ? 0 : (mlp == 1) ? 32 : 64;
      const int Ks  = (mlp == 2) ? 16 : 8;
      const int Bnt = (mlp == 2) ? 4 : 1;
      const v2f* wB = (const v2f*)(sm + ((mlp == 0) ? OFF_W3E : (mlp == 1) ? OFF_W3P : OFF_W3V));
      v8f c = {};
      for (int ks = 0; ks < Ks; ++ks) {
        v2f b = wB[(ks * Bnt + nt) * 32 + lane];
        v2f a = *(const v2f*)(sm + OFF_H2 + (mt * 16 + arow) * H1S + Ain + ks * 4 + koff);
        c = wmma4(a, b, c);
      }
      const int cc = nt * 16 + arow;
      const int rb = (lane >> 4) * 8;
      const float bias = sm[OFF_B3 + ((mlp == 0) ? 0 : (mlp == 1) ? 16 : 32) + cc];
      const int   ob   = (mlp == 0) ? OFF_KQ : (mlp == 1) ? OFF_EPK : OFF_V;
      const int   strd = (mlp == 2) ? 68 : 16;
#pragma unroll
      for (int v = 0; v < 8; ++v)
        sm[ob + (mt * 16 + rb + v) * strd + cc] = c[v] + bias;  // no activation
    }
  }
  __syncthreads();

  // ================= Phase F: attention logits + softmax ====================
  if (tid < 128) {                         // Mq logits: (2 heads x 64 keys)
    int h = tid >> 6, e = tid & 63;
    const float* Q0 = sm + OFF_Q + h * 3;
    float l = (Q0[0] * sm[OFF_KQ + e * 16 + 0] +
               Q0[1] * sm[OFF_KQ + e * 16 + 1] +
               Q0[2] * sm[OFF_KQ + e * 16 + 2]) / SDK;
    sm[OFF_MQ + h * 64 + e] = l;
  }
  for (int i = tid; i < 384; i += 256) {   // Mp logits: (2 heads x 192 keys)
    int h = i / 192, j = i % 192;
    int kk = j >> 6, e = j & 63;
    const float* Q1 = sm + OFF_Q + 6 + h * 3;
    const float* ep = sm + OFF_EPK + e * 16 + kk * 3;
    sm[OFF_MP + h * 192 + j] = (Q1[0] * ep[0] + Q1[1] * ep[1] + Q1[2] * ep[2]) / SDK;
  }
  __syncthreads();
  if (tid < 2) {                           // softmax over 64 (per head)
    float* L = sm + OFF_MQ + tid * 64;
    float m = L[0];
    for (int e = 1; e < 64; ++e) m = fmaxf(m, L[e]);
    float s = 0.0f;
    for (int e = 0; e < 64; ++e) { float x = expf(L[e] - m); L[e] = x; s += x; }
    float inv = 1.0f / s;
    for (int e = 0; e < 64; ++e) L[e] *= inv;
  } else if (tid < 4) {                    // softmax over 192 (per head)
    int h = tid - 2;
    float* L = sm + OFF_MP + h * 192;
    float m = L[0];
    for (int j = 1; j < 192; ++j) m = fmaxf(m, L[j]);
    float s = 0.0f;
    for (int j = 0; j < 192; ++j) { float x = expf(L[j] - m); L[j] = x; s += x; }
    float inv = 1.0f / s, mid = 0.0f;
    for (int j = 0; j < 192; ++j) { L[j] *= inv; if (j >= 64 && j < 128) mid += L[j]; }
    sm[OFF_MPS + h] = mid;                 // sum of middle block (p broadcast term)
  }
  __syncthreads();

  // ================= Phase G: Zq and Zp (all-LDS) ===========================
  if (tid < 64) {                          // Zq[h*32+s] = Mq[h] . V0[:,s]
    int h = tid >> 5, s = tid & 31;
    float acc = 0.0f;
    for (int e = 0; e < 64; ++e) acc += sm[OFF_MQ + h * 64 + e] * sm[OFF_V + e * 68 + s];
    sm[OFF_ZQ + tid] = acc;
  } else {                                 // Zp[d][h*32+s], 192 items
    int i = tid - 64;
    int d = i >> 6, c = i & 63, h = c >> 5, s = c & 31;
    const float* M = sm + OFF_MP + h * 192;
    float acc = sm[OFF_MPS + h] * sm[OFF_PS + d * 32 + s];
    for (int e = 0; e < 64; ++e) {
      acc += M[e]       * sm[OFF_V + e * 68 + 32 + s] * sm[OFF_RS + e * 3 + d];
      acc += M[128 + e] * sm[OFF_PN + (e * 3 + d) * NSZ + s];
    }
    sm[OFF_ZP + i] = acc;
  }
  __syncthreads();

  // ================= Phase H: qpm MLP + ph ==================================
  if (tid < 32) {
    float acc = P.qpB0[tid];
    for (int k = 0; k < 64; ++k) acc += sm[OFF_ZQ + k] * P.qpW0[k * 32 + tid];
    sm[OFF_TMP + tid] = lrelu(acc);
  } else if (tid < 128) {                  // ph = Zp(3x64) @ ppm_W(64x32)
    int i = tid - 32, d = i >> 5, s = i & 31;
    float acc = 0.0f;
    for (int c = 0; c < 64; ++c) acc += sm[OFF_ZP + d * 64 + c] * P.ppW[c * 32 + s];
    sm[OFF_PH + i] = acc;
  }
  __syncthreads();
  if (tid < 32) {
    float acc = P.qpB1[tid];
    for (int k = 0; k < 32; ++k) acc += sm[OFF_TMP + k] * P.qpW1[k * 32 + tid];
    sm[OFF_TMP + 32 + tid] = lrelu(acc);
  }
  __syncthreads();
  if (tid < 32) {
    float acc = P.qpB2[tid];
    for (int k = 0; k < 32; ++k) acc += sm[OFF_TMP + 32 + k] * P.qpW2[k * 32 + tid];
    sm[OFF_QH + tid] = acc;
  }
  __syncthreads();

  // ================= Phase I: residual outputs ==============================
  if (tid < 32) P.out_q[(size_t)node * 32 + tid] = qg[tid] + sm[OFF_QH + tid];
  if (tid >= 32 && tid < 128) {
    int i = tid - 32;
    P.out_p[(size_t)node * 96 + i] = pg[i] + sm[OFF_PH + i];
  }
}

extern "C" void kernel_launch(void* const* d_in, const int* in_sizes, int n_in,
                              void* d_out, int out_size, void* d_ws, size_t ws_size,
                              hipStream_t stream) {
  (void)n_in; (void)out_size; (void)d_ws; (void)ws_size;
  Params P;
  P.q    = (const float*)d_in[0];
  P.p    = (const float*)d_in[1];
  P.q_nn = (const float*)d_in[2];
  P.p_nn = (const float*)d_in[3];
  P.d_nn = (const float*)d_in[4];
  P.r_nn = (const float*)d_in[5];
  P.nqW0 = (const float*)d_in[6];  P.nqB0 = (const float*)d_in[7];
  P.nqW1 = (const float*)d_in[8];  P.nqB1 = (const float*)d_in[9];
  P.nqW2 = (const float*)d_in[10]; P.nqB2 = (const float*)d_in[11];
  P.eqW0 = (const float*)d_in[12]; P.eqB0 = (const float*)d_in[13];
  P.eqW1 = (const float*)d_in[14]; P.eqB1 = (const float*)d_in[15];
  P.eqW2 = (const float*)d_in[16]; P.eqB2 = (const float*)d_in[17];
  P.epW0 = (const float*)d_in[18]; P.epB0 = (const float*)d_in[19];
  P.epW1 = (const float*)d_in[20]; P.epB1 = (const float*)d_in[21];
  P.epW2 = (const float*)d_in[22]; P.epB2 = (const float*)d_in[23];
  P.evW0 = (const float*)d_in[24]; P.evB0 = (const float*)d_in[25];
  P.evW1 = (const float*)d_in[26]; P.evB1 = (const float*)d_in[27];
  P.evW2 = (const float*)d_in[28]; P.evB2 = (const float*)d_in[29];
  P.qpW0 = (const float*)d_in[30]; P.qpB0 = (const float*)d_in[31];
  P.qpW1 = (const float*)d_in[32]; P.qpB1 = (const float*)d_in[33];
  P.qpW2 = (const float*)d_in[34]; P.qpB2 = (const float*)d_in[35];
  P.ppW  = (const float*)d_in[36];
  P.N    = in_sizes[0] / NSZ;                // 8192 nodes
  P.out_q = (float*)d_out;
  P.out_p = (float*)d_out + (size_t)P.N * NSZ;

  const size_t smem = SM_FLOATS * sizeof(float);   // ~313 KB (<320 KB/WGP)
  (void)hipFuncSetAttribute((const void*)state_update_kernel,
                            hipFuncAttributeMaxDynamicSharedMemorySize, (int)smem);
  state_update_kernel<<<dim3((unsigned)P.N), dim3(256), smem, stream>>>(P);
}